// FastEWRLSLevel_4870492914165
// MI455X (gfx1250) — compile-verified
//
#include <hip/hip_runtime.h>
#include <stdint.h>

// EWRLS level filter, 3-phase chunked linear-recurrence scan for gfx1250 (MI455X).
//
// Roofline: B*T*C = 67M elements, ~6 VALU ops each (<1 GFLOP) vs 768 MB of HBM
// traffic (x read twice + out written once) => ~33 us at 23.3 TB/s. Memory-bound;
// WMMA is not applicable. The CDNA5 features that matter are the DMA engines:
//   Phase 1: Tensor Data Mover (tensor_load_to_lds + s_wait_tensorcnt), one 2D
//            descriptor per 16x256 tile, issued by wave 0 (TDM ignores EXEC).
//   Phase 3: per-lane async DMA (global_load_async_to_lds_b128 + s_wait_asynccnt),
//            double-buffered in LDS.
// K is recomputed on the fly from the closed form P_t = 1/(lam^t + (1-lam^t)/(1-lam));
// no K table in memory. Workspace: 3 * 32*32*512 floats = 6 MB.

#define B_ 32
#define T_ 4096
#define C_ 512
#define L_ 128            // chunk length (timesteps)
#define NC_ (T_ / L_)     // 32 chunks
#define TT_ 16            // timesteps per LDS tile
#define NTILES_ (L_ / TT_)
#define CLAMP_LO 1.0e-4f
#define CLAMP_HI (1.0f - 1.0e-4f)

typedef uint32_t v4u __attribute__((ext_vector_type(4)));
typedef int      v4i __attribute__((ext_vector_type(4)));
typedef int      v8i __attribute__((ext_vector_type(8)));

#define S_WAIT_ASYNCCNT(N) asm volatile("s_wait_asynccnt " #N ::: "memory")

__device__ __forceinline__ void async_ld_b128(uint32_t lds_addr, const float* gp) {
  // CDNA5 async DMA: global -> LDS, tracked by ASYNCcnt (no VGPR destination).
  asm volatile("global_load_async_to_lds_b128 %0, %1, off"
               :: "v"(lds_addr), "v"(gp) : "memory");
}

// Tensor Data Mover: load one 16-row x 256-float tile (row stride C_ floats)
// from global into LDS. D# built per CDNA5 ISA ch.8 (group0 + group1, 2D tensor).
__device__ __forceinline__ void tdm_load_tile(uint32_t lds_addr, const float* gp) {
  uint64_t ga = (uint64_t)(uintptr_t)gp;
  v4u g0;
  g0.x = 1u;                                               // count=1, user D#, no gather
  g0.y = lds_addr;                                         // LDS byte address
  g0.z = (uint32_t)ga;                                     // global_addr[31:0]
  g0.w = (uint32_t)((ga >> 32) & 0x01FFFFFFu) | (2u << 30); // global_addr[56:32] | type=2
  v8i g1;
  g1[0] = (int)(2u << 16);          // workgroup_mask=0 | data_size=2 (4 bytes)
  g1[1] = (int)(256u << 16);        // atomic_barrier_addr=0 | tensor_dim0.lo16=256
  g1[2] = (int)((uint32_t)TT_ << 16);   // tensor_dim0.hi16=0 | tensor_dim1.lo16=16
  g1[3] = (int)(256u << 16);        // tensor_dim1.hi16=0 | tile_dim0=256
  g1[4] = TT_;                      // tile_dim1=16 | tile_dim2=0
  g1[5] = C_;                       // tensor_dim0_stride.lo32 = 512 elements
  g1[6] = 0;                        // stride0.hi16=0 | stride1.lo16=0
  g1[7] = 0;                        // stride1.hi32=0
  v4i z4 = {0, 0, 0, 0};
#if __has_include(<hip/amd_detail/amd_gfx1250_TDM.h>)
  v8i z8 = {0, 0, 0, 0, 0, 0, 0, 0};
  __builtin_amdgcn_tensor_load_to_lds(g0, g1, z4, z4, z8, 0);  // clang-23 6-arg form
#else
  __builtin_amdgcn_tensor_load_to_lds(g0, g1, z4, z4, 0);      // ROCm 7.2 5-arg form
#endif
}

__device__ __forceinline__ float lam_of(const float* __restrict__ ll, int c) {
  float z = ll[c];
  float lam = 1.0f / (1.0f + __expf(-z));
  return fminf(fmaxf(lam, CLAMP_LO), CLAMP_HI);
}

// Carry P before step t+1:  Q_t = 1 / (lam^t + (1 - lam^t) / (1 - lam)),  Q_0 = 1.
// K_t (gain applied at step t) == Q_t for t >= 1.
__device__ __forceinline__ float q_closed(float lam, int t) {
  float lt = __expf((float)t * __logf(lam));   // lam^t
  return 1.0f / (lt + (1.0f - lt) / (1.0f - lam));
}

// ---------------- Phase 1: chunk summaries, x staged via TDM ----------------
extern "C" __global__ __launch_bounds__(256)
void ewrls_phase1(const float* __restrict__ x, const float* __restrict__ ll,
                  float* __restrict__ sumA, float* __restrict__ sumB) {
  __shared__ float tile[2][TT_][256];           // 2 x 16KB double buffer

  int tid = threadIdx.x;
  int blk = blockIdx.x;                         // (j, b, c-half)
  int c0 = (blk & 1) * 256;
  int b  = (blk >> 1) % B_;
  int j  = blk / (2 * B_);
  int c  = c0 + tid;
  int t0 = j * L_;

  float lam = lam_of(ll, c);
  float q  = (j == 0) ? 1.0f : q_closed(lam, t0 - 1);
  float th = 0.0f;                              // overwritten at t==0 for chunk 0
  float A  = 1.0f;

  const float* gtile0 = x + ((size_t)b * T_ + t0) * C_ + c0;
  uint32_t lds0 = (uint32_t)(uintptr_t)&tile[0][0][0];  // low 32 bits = LDS byte offset

  if (tid == 0) tdm_load_tile(lds0, gtile0);    // prime buffer 0 (TDM ignores EXEC)

  int bi = 0;
  for (int ti = 0; ti < NTILES_; ++ti) {
    if (tid == 0) {
      if (ti + 1 < NTILES_) {
        tdm_load_tile(lds0 + (uint32_t)((bi ^ 1) * (TT_ * 256 * 4)),
                      gtile0 + (size_t)(ti + 1) * TT_ * C_);
        __builtin_amdgcn_s_wait_tensorcnt(1);   // only the just-issued tile pending
      } else {
        __builtin_amdgcn_s_wait_tensorcnt(0);
      }
    }
    __syncthreads();                            // tile ti visible to all 8 waves

    int tbase = t0 + ti * TT_;
    for (int s = 0; s < TT_; ++s) {
      float xv = tile[bi][s][tid];
      int t = tbase + s;
      if (t == 0) {
        th = xv;                                // theta_0 = x_0 (chunk 0 only)
      } else {
        q = __fdividef(q, lam + q);             // K_t
        float a = 1.0f - q;
        A *= a;
        th = a * th + q * xv;
      }
    }
    __syncthreads();                            // done reading buf bi before refill
    bi ^= 1;
  }

  size_t w = ((size_t)j * B_ + b) * C_ + c;
  sumA[w] = (j == 0) ? 0.0f : A;                // chunk 0 has no incoming carry
  sumB[w] = th;
}

// ---------------- Phase 2: scan the 32 chunk summaries ----------------
extern "C" __global__ __launch_bounds__(256)
void ewrls_phase2(const float* __restrict__ sumA, const float* __restrict__ sumB,
                  float* __restrict__ carry) {
  int flat = blockIdx.x * 256 + threadIdx.x;    // (b, c)
  int c = flat % C_;
  int b = flat / C_;
  float h = 0.0f;                               // theta entering chunk j
  for (int j = 0; j < NC_; ++j) {
    size_t w = ((size_t)j * B_ + b) * C_ + c;
    carry[w] = h;
    h = sumA[w] * h + sumB[w];
  }
}

// ---------------- Phase 3: replay chunks, x staged via per-lane async DMA ----------------
extern "C" __global__ __launch_bounds__(256)
void ewrls_phase3(const float* __restrict__ x, const float* __restrict__ ll,
                  const float* __restrict__ carry, float* __restrict__ out) {
  __shared__ float tile[2][TT_][256];           // 2 x 16KB double buffer

  int tid = threadIdx.x;
  int blk = blockIdx.x;                         // (j, b, c-half)
  int c0 = (blk & 1) * 256;
  int b  = (blk >> 1) % B_;
  int j  = blk / (2 * B_);
  int c  = c0 + tid;
  int t0 = j * L_;

  float lam = lam_of(ll, c);
  float q = (j == 0) ? 1.0f : q_closed(lam, t0 - 1);
  float h = carry[((size_t)j * B_ + b) * C_ + c];

  const float* gtile0 = x + ((size_t)b * T_ + t0) * C_ + c0;  // 256-float rows, stride C_
  uint32_t lds0 = (uint32_t)(uintptr_t)&tile[0][0][0];
  int lrow = tid >> 6;                          // 4 rows per b128 issue group
  int lcol = (tid & 63) << 2;                   // float index within row (16B per lane)

  // Prime buffer 0 with tile 0 (4 async b128 per thread = one 16x256 tile per block).
  for (int g = 0; g < 4; ++g) {
    int r = g * 4 + lrow;
    async_ld_b128(lds0 + (uint32_t)(r * 1024 + lcol * 4),
                  gtile0 + (size_t)r * C_ + lcol);
  }

  float* op = out + (size_t)b * T_ * C_ + c;
  int bi = 0;
  for (int ti = 0; ti < NTILES_; ++ti) {
    if (ti + 1 < NTILES_) {
      const float* gb = gtile0 + (size_t)(ti + 1) * TT_ * C_;
      uint32_t lb = lds0 + (uint32_t)((bi ^ 1) * (TT_ * 256 * 4));
      for (int g = 0; g < 4; ++g) {
        int r = g * 4 + lrow;
        async_ld_b128(lb + (uint32_t)(r * 1024 + lcol * 4),
                      gb + (size_t)r * C_ + lcol);
      }
      S_WAIT_ASYNCCNT(4);     // <=4 outstanding: only the just-issued next tile
    } else {
      S_WAIT_ASYNCCNT(0);
    }
    __syncthreads();          // all waves' DMA for this tile is visible in LDS

    int tbase = t0 + ti * TT_;
    for (int s = 0; s < TT_; ++s) {
      float xv = tile[bi][s][tid];
      int t = tbase + s;
      if (t == 0) {
        h = xv;                               // theta_0 = x_0
      } else {
        q = __fdividef(q, lam + q);           // K_t
        h = h + q * (xv - h);
      }
      // Streaming write-once output: non-temporal, keep L2 for x.
      __builtin_nontemporal_store(h, &op[(size_t)t * C_]);
    }
    __syncthreads();          // everyone done reading buf bi before it is re-filled
    bi ^= 1;
  }
}

extern "C" void kernel_launch(void* const* d_in, const int* in_sizes, int n_in,
                              void* d_out, int out_size, void* d_ws, size_t ws_size,
                              hipStream_t stream) {
  const float* x  = (const float*)d_in[0];   // [32, 4096, 512] f32
  const float* ll = (const float*)d_in[1];   // [512] f32
  float* out = (float*)d_out;                // [32, 4096, 512] f32

  const size_t nsum = (size_t)NC_ * B_ * C_; // 524288
  float* sumA  = (float*)d_ws;               // 2 MB
  float* sumB  = sumA + nsum;                // 2 MB
  float* carry = sumB + nsum;                // 2 MB   (6 MB total of d_ws)

  const int blocks = NC_ * B_ * (C_ / 256);  // 2048
  ewrls_phase1<<<blocks, 256, 0, stream>>>(x, ll, sumA, sumB);
  ewrls_phase2<<<(B_ * C_) / 256, 256, 0, stream>>>(sumA, sumB, carry);
  ewrls_phase3<<<blocks, 256, 0, stream>>>(x, ll, carry, out);
}